// Net_55224689492388
// MI455X (gfx1250) — compile-verified
//
#include <hip/hip_runtime.h>
#include <hip/hip_bf16.h>
#include <stdint.h>

// Problem constants (from reference): N=65536 sequences, T=128 steps, D=2 channels.
#define BLK     128    // threads per block = sequences per block (4 waves)
#define TSTEPS  128
#define DDIM    2
#define LDS_T   64     // timesteps staged in LDS (the burn-in window)
#define ROWD    132    // dwords per sequence row in LDS: 64*2 data + 4 pad (528B, 16B aligned)

// ---- CDNA5 async global->LDS copy (ASYNCcnt path), inline asm for toolchain portability ----
__device__ __forceinline__ void async_ld_b128(uint32_t lds_byte_off, const void* gaddr) {
    // VDST = per-lane LDS byte address, VADDR = per-lane 64-bit global address
    asm volatile("global_load_async_to_lds_b128 %0, %1, off"
                 :: "v"(lds_byte_off), "v"(gaddr) : "memory");
}
__device__ __forceinline__ void wait_async0() {
    asm volatile("s_wait_asynccnt 0" ::: "memory");
}

__global__ __launch_bounds__(BLK, 2) void inertia_scan_kernel(
    const float* __restrict__ src, const float* __restrict__ msk,
    const int* __restrict__ burn_p, float* __restrict__ out, int N)
{
    __shared__ __align__(16) float s_src[BLK * ROWD];
    __shared__ __align__(16) float s_msk[BLK * ROWD];

    const int tid  = threadIdx.x;
    const int lane = tid & 31;
    const int wave = tid >> 5;
    const int n0   = blockIdx.x * BLK;
    const int burn = burn_p[0];

    const uint32_t lds_src_base = (uint32_t)(uintptr_t)(void*)s_src;
    const uint32_t lds_msk_base = (uint32_t)(uintptr_t)(void*)s_msk;

    // Stage the burn-in window (first 64 timesteps = 512B per sequence) of source
    // and mask into LDS, wave-locally. Async instruction j copies sequence
    // (n0 + wave*32 + j): its 32 lanes fetch 32 x 16B = 512B *contiguous* global
    // bytes (fully coalesced) into that sequence's padded LDS row.
    {
        const int nb = n0 + wave * 32;
        const float* gs = src + (size_t)nb * (TSTEPS * DDIM) + lane * 4;
        const float* gm = msk + (size_t)nb * (TSTEPS * DDIM) + lane * 4;
        uint32_t ls = lds_src_base + (uint32_t)(wave * 32) * (ROWD * 4) + lane * 16;
        uint32_t lm = lds_msk_base + (uint32_t)(wave * 32) * (ROWD * 4) + lane * 16;
        for (int j = 0; j < 32; ++j) {
            if (nb + j < N) {
                async_ld_b128(ls, gs);
                async_ld_b128(lm, gm);
            }
            gs += TSTEPS * DDIM;    // next sequence, same lane slot
            gm += TSTEPS * DDIM;
            ls += ROWD * 4;
            lm += ROWD * 4;
        }
    }
    wait_async0();   // wave-local producers/consumers: no workgroup barrier needed

    const int n = n0 + tid;
    if (n >= N) return;

    const float* __restrict__ srow = s_src + tid * ROWD;
    const float* __restrict__ mrow = s_msk + tid * ROWD;
    const size_t gbase = (size_t)n * (TSTEPS * DDIM);

    // Effective teacher-forced extent: burn<=0 means "always in burn-in".
    const int be    = (burn <= 0 || burn > TSTEPS) ? TSTEPS : burn;  // [0,be) forced
    const int p1end = (be < LDS_T) ? be : LDS_T;                     // LDS-covered part

    // Recurrence state per channel: p = s[d], v = s[2+d], mm = mask carry, yy = y carry.
    float p0 = 0.f, p1 = 0.f, v0 = 0.f, v1 = 0.f;
    float mm0 = 0.f, mm1 = 0.f, yy0 = 0.f, yy1 = 0.f;
    float4 ob;
    int t = 0;

    // ---- Phase 1: teacher-forced, inputs from LDS ----
    #pragma unroll 8
    for (; t < p1end; ++t) {
        const float2 sx = *(const float2*)(srow + 2 * t);
        const float2 mx = *(const float2*)(mrow + 2 * t);
        const float x0 = sx.x, x1 = sx.y, k0 = mx.x, k1 = mx.y;
        const float app0 = (1.f - mm0) * k0;          // s_app = (1-m_prev)*m_new
        const float app1 = (1.f - mm1) * k1;
        const float vt0 = x0 - p0, vt1 = x1 - p1;     // s_temp[2:4] = x - s[0:2]
        v0 = vt0 + app0 * (v0 - vt0);                 // = v*app + vt*(1-app)
        v1 = vt1 + app1 * (v1 - vt1);
        p0 = x0; p1 = x1;
        yy0 = x0 + v0; yy1 = x1 + v1;
        mm0 = k0; mm1 = k1;
        if (t & 1) { ob.z = yy0; ob.w = yy1; *(float4*)(out + gbase + 2 * (t - 1)) = ob; }
        else       { ob.x = yy0; ob.y = yy1; }
    }

    // ---- Phase 2 (rare: burn>64 or burn<=0): teacher-forced from global ----
    for (; t < be; ++t) {
        const float x0 = src[gbase + 2 * t], x1 = src[gbase + 2 * t + 1];
        const float k0 = msk[gbase + 2 * t], k1 = msk[gbase + 2 * t + 1];
        const float app0 = (1.f - mm0) * k0;
        const float app1 = (1.f - mm1) * k1;
        const float vt0 = x0 - p0, vt1 = x1 - p1;
        v0 = vt0 + app0 * (v0 - vt0);
        v1 = vt1 + app1 * (v1 - vt1);
        p0 = x0; p1 = x1;
        yy0 = x0 + v0; yy1 = x1 + v1;
        mm0 = k0; mm1 = k1;
        if (t & 1) { ob.z = yy0; ob.w = yy1; *(float4*)(out + gbase + 2 * (t - 1)) = ob; }
        else       { ob.x = yy0; ob.y = yy1; }
    }

    // ---- Phase 3: autoregressive. m frozen => s_app loop-invariant; no loads. ----
    const float app0 = (1.f - mm0) * mm0;
    const float app1 = (1.f - mm1) * mm1;
    #pragma unroll 8
    for (; t < TSTEPS; ++t) {
        const float x0 = yy0, x1 = yy1;               // x = previous y
        const float vt0 = x0 - p0, vt1 = x1 - p1;
        v0 = vt0 + app0 * (v0 - vt0);
        v1 = vt1 + app1 * (v1 - vt1);
        p0 = x0; p1 = x1;
        yy0 = x0 + v0; yy1 = x1 + v1;
        if (t & 1) { ob.z = yy0; ob.w = yy1; *(float4*)(out + gbase + 2 * (t - 1)) = ob; }
        else       { ob.x = yy0; ob.y = yy1; }
    }
}

extern "C" void kernel_launch(void* const* d_in, const int* in_sizes, int n_in,
                              void* d_out, int out_size, void* d_ws, size_t ws_size,
                              hipStream_t stream) {
    // setup_inputs order: source, mask, A, B, C, burn_in_steps
    const float* src   = (const float*)d_in[0];
    const float* msk   = (const float*)d_in[1];
    const int*   burnp = (const int*)d_in[5];   // scalar -> single-element array
    float*       out   = (float*)d_out;

    const int N = in_sizes[0] / (TSTEPS * DDIM);
    const int grid = (N + BLK - 1) / BLK;
    hipLaunchKernelGGL(inertia_scan_kernel, dim3(grid), dim3(BLK), 0, stream,
                       src, msk, burnp, out, N);
}